// CoEvolutionBlock_76957224009945
// MI455X (gfx1250) — compile-verified
//
#include <hip/hip_runtime.h>
#include <hip/hip_bf16.h>

#define BB 2
#define LS 256
#define SD 64
#define PD 32
#define NH 4
#define HD 16
#define OH 32
#define EXPC 64
#define SEC 16
#define LL 65536

typedef __attribute__((ext_vector_type(2))) float v2f;
typedef __attribute__((ext_vector_type(8))) float v8f;

__device__ inline float blockReduceSum256(float v, float* sm) {
    int t = threadIdx.x;
    sm[t] = v; __syncthreads();
    for (int s = 128; s > 0; s >>= 1) { if (t < s) sm[t] += sm[t + s]; __syncthreads(); }
    float r = sm[0]; __syncthreads();
    return r;
}

__device__ inline float siluf(float x) { return x / (1.f + __expf(-x)); }

// ---------------- K1: seq LN -> x, a = x@W_opm_lin, per-row stats of a ----------------
__global__ void k_seq_ln_a(const float* __restrict__ seq, const float* __restrict__ g_attn,
                           const float* __restrict__ b_attn, const float* __restrict__ W_lin,
                           const float* __restrict__ b_lin, float* __restrict__ xbuf,
                           float* __restrict__ abuf, float* __restrict__ mstat,
                           float* __restrict__ sstat) {
    __shared__ float sm[64], xs[64], asv[32];
    int row = blockIdx.x, t = threadIdx.x;
    float v = seq[row * 64 + t];
    sm[t] = v; __syncthreads();
    for (int s = 32; s > 0; s >>= 1) { if (t < s) sm[t] += sm[t + s]; __syncthreads(); }
    float mean = sm[0] / 64.f; __syncthreads();
    float d = v - mean;
    sm[t] = d * d; __syncthreads();
    for (int s = 32; s > 0; s >>= 1) { if (t < s) sm[t] += sm[t + s]; __syncthreads(); }
    float inv = rsqrtf(sm[0] / 64.f + 1e-5f); __syncthreads();
    float xv = d * inv * g_attn[t] + b_attn[t];
    xs[t] = xv;
    xbuf[row * 64 + t] = xv;
    __syncthreads();
    if (t < 32) {
        float acc = b_lin[t];
        for (int k = 0; k < 64; k++) acc += xs[k] * W_lin[k * 32 + t];
        abuf[row * 32 + t] = acc;
        asv[t] = acc;
    }
    __syncthreads();
    if (t == 0) {
        float s1 = 0.f, s2 = 0.f;
        for (int p = 0; p < 32; p++) { float av = asv[p]; s1 += av; s2 += av * av; }
        mstat[row] = s1 / 32.f; sstat[row] = s2 / 32.f;
    }
}

// ---------------- K2: G[d,e,p] = g_opm[de]*Wproj[de,p]; C1/C2 constants ----------------
__global__ void k_prep_G(const float* __restrict__ g_opm, const float* __restrict__ b_opm,
                         const float* __restrict__ Wp, const float* __restrict__ bp,
                         float* __restrict__ G, float* __restrict__ C1, float* __restrict__ C2) {
    int idx = blockIdx.x * 256 + threadIdx.x;
    if (blockIdx.x < 128) {
        int de = idx >> 5, p = idx & 31;
        G[idx] = g_opm[de] * Wp[de * 32 + p];
    } else if (threadIdx.x < 32) {
        int p = threadIdx.x;
        float c1 = 0.f, c2 = 0.f;
        for (int de = 0; de < 1024; de++) {
            float w = Wp[de * 32 + p];
            c1 += g_opm[de] * w; c2 += b_opm[de] * w;
        }
        C1[p] = c1; C2[p] = c2 + bp[p];
    }
}

// ---------------- K3: Y2[b][d][j*32+p] = sum_e a[b,j,e]*G[d,e,p] ----------------
__global__ void k_Y(const float* __restrict__ abuf, const float* __restrict__ G,
                    float* __restrict__ Y2) {
    __shared__ float as[32];
    int b = blockIdx.x >> 8, j = blockIdx.x & 255;
    int t = threadIdx.x; // 1024
    if (t < 32) as[t] = abuf[(b * 256 + j) * 32 + t];
    __syncthreads();
    int d = t >> 5, p = t & 31;
    float acc = 0.f;
    for (int e = 0; e < 32; e++) acc += as[e] * G[d * 1024 + e * 32 + p];
    Y2[(size_t)b * 262144 + d * 8192 + j * 32 + p] = acc;
}

// ---------------- K4: WMMA GEMM [256x32]@[32x8192] + analytic-LN + pair residual ----------------
__global__ void k_opm_wmma(const float* __restrict__ abuf, const float* __restrict__ Y2,
                           const float* __restrict__ mstat, const float* __restrict__ sstat,
                           const float* __restrict__ C1, const float* __restrict__ C2,
                           const float* __restrict__ pair_repr, float* __restrict__ pair1) {
    int wave = (blockIdx.x << 2) | (threadIdx.x >> 5);
    int lane = threadIdx.x & 31;
    int half = lane >> 4, lr = lane & 15;
    int b = wave >> 13;           // 8192 tiles per batch (16 x 512)
    int rem = wave & 8191;
    int m0 = (rem >> 9) << 4;
    int n0 = (rem & 511) << 4;
    const float* A = abuf + b * 256 * 32;
    const float* Bm = Y2 + (size_t)b * 262144;
    int arow = m0 + lr;
    int bcol = n0 + lr;
    v8f c = {};
#pragma unroll
    for (int kk = 0; kk < 8; kk++) {
        int k0 = kk * 4 + half * 2;
        v2f av; av.x = A[arow * 32 + k0]; av.y = A[arow * 32 + k0 + 1];
        v2f bv; bv.x = Bm[(size_t)k0 * 8192 + bcol]; bv.y = Bm[(size_t)(k0 + 1) * 8192 + bcol];
        c = __builtin_amdgcn_wmma_f32_16x16x4_f32(false, av, false, bv, (short)0, c, false, false);
    }
    int j = bcol >> 5, p = bcol & 31;
    float mj = mstat[b * 256 + j], sj = sstat[b * 256 + j];
    float c1p = C1[p], c2p = C2[p];
#pragma unroll
    for (int r = 0; r < 8; r++) {
        int i = m0 + r + half * 8;
        float mi = mstat[b * 256 + i], si = sstat[b * 256 + i];
        float m = mi * mj;
        float inv = rsqrtf(si * sj - m * m + 1e-5f);
        size_t idx = (((size_t)(b * 256 + i)) * 256 + j) * 32 + p;
        pair1[idx] = c[r] * inv - m * inv * c1p + c2p + pair_repr[idx];
    }
}

// ---------------- K5: pn = LN(pair1) over 32 channels ----------------
__global__ void k_pn(const float* __restrict__ pair1, const float* __restrict__ g_pair,
                     const float* __restrict__ b_pair, float* __restrict__ pn) {
    __shared__ float gp[32], bp[32];
    int t = threadIdx.x;
    if (t < 32) { gp[t] = g_pair[t]; bp[t] = b_pair[t]; }
    __syncthreads();
    size_t base = ((size_t)blockIdx.x * 256 + t) * 32;
    float v[32]; float s1 = 0.f;
#pragma unroll
    for (int p = 0; p < 32; p++) { v[p] = pair1[base + p]; s1 += v[p]; }
    float m = s1 / 32.f, s2 = 0.f;
#pragma unroll
    for (int p = 0; p < 32; p++) { float d = v[p] - m; s2 += d * d; }
    float inv = rsqrtf(s2 / 32.f + 1e-5f);
#pragma unroll
    for (int p = 0; p < 32; p++) pn[base + p] = (v[p] - m) * inv * gp[p] + bp[p];
}

// ---------------- K6: expand conv1x1 (32->64) + per-(b,c) partial sums ----------------
__global__ void k_e(const float* __restrict__ pn, const float* __restrict__ W_exp,
                    float* __restrict__ E, float* __restrict__ part) {
    __shared__ float w[32]; __shared__ float sm[256];
    int blk = blockIdx.x;
    int i = blk & 255, c = (blk >> 8) & 63, b = blk >> 14;
    int j = threadIdx.x;
    if (j < 32) w[j] = W_exp[c * 32 + j];
    __syncthreads();
    const float* row = pn + (((size_t)(b * 256 + i)) * 256 + j) * 32;
    float acc = 0.f;
#pragma unroll
    for (int p = 0; p < 32; p++) acc += row[p] * w[p];
    E[((size_t)(b * 64 + c)) * LL + i * 256 + j] = acc;
    float s1 = blockReduceSum256(acc, sm);
    float s2 = blockReduceSum256(acc * acc, sm);
    if (j == 0) { part[((b * 64 + c) * 256 + i) * 2] = s1; part[((b * 64 + c) * 256 + i) * 2 + 1] = s2; }
}

// ---------------- generic: reduce 256 (sum,sumsq) partials -> (mean, invstd) ----------------
__global__ void k_reduce_stats(const float* __restrict__ part, float* __restrict__ stats,
                               float inv_count) {
    __shared__ float sm[256];
    int g = blockIdx.x, t = threadIdx.x;
    float s1 = part[(g * 256 + t) * 2], s2 = part[(g * 256 + t) * 2 + 1];
    s1 = blockReduceSum256(s1, sm);
    s2 = blockReduceSum256(s2, sm);
    if (t == 0) {
        float m = s1 * inv_count;
        float var = s2 * inv_count - m * m;
        stats[g * 2] = m; stats[g * 2 + 1] = rsqrtf(var + 1e-5f);
    }
}

// ---------------- elementwise silu(instancenorm(x)) in place, [G][65536] ----------------
__global__ void k_silu_norm(float* __restrict__ X, const float* __restrict__ stats) {
    size_t idx = (size_t)blockIdx.x * 256 + threadIdx.x;
    int g = (int)(idx >> 16);
    float v = (X[idx] - stats[g * 2]) * stats[g * 2 + 1];
    X[idx] = siluf(v);
}

// ---------------- depthwise 3x3 SAME + partial sums ----------------
__global__ void k_dw(const float* __restrict__ HS, const float* __restrict__ W_dw,
                     float* __restrict__ D, float* __restrict__ part) {
    __shared__ float w[9]; __shared__ float sm[256];
    int blk = blockIdx.x;
    int i = blk & 255, c = (blk >> 8) & 63, b = blk >> 14;
    int j = threadIdx.x;
    if (j < 9) w[j] = W_dw[c * 9 + j];
    __syncthreads();
    const float* src = HS + ((size_t)(b * 64 + c)) * LL;
    float acc = 0.f;
#pragma unroll
    for (int u = 0; u < 3; u++) {
        int ii = i + u - 1;
        if (ii < 0 || ii > 255) continue;
#pragma unroll
        for (int vv = 0; vv < 3; vv++) {
            int jj = j + vv - 1;
            if (jj < 0 || jj > 255) continue;
            acc += src[ii * 256 + jj] * w[u * 3 + vv];
        }
    }
    D[((size_t)(b * 64 + c)) * LL + i * 256 + j] = acc;
    float s1 = blockReduceSum256(acc, sm);
    float s2 = blockReduceSum256(acc * acc, sm);
    if (j == 0) { part[((b * 64 + c) * 256 + i) * 2] = s1; part[((b * 64 + c) * 256 + i) * 2 + 1] = s2; }
}

// ---------------- h2 = silu(inorm(d)) in place + sum partials (for SE squeeze) ----------------
__global__ void k_h2(float* __restrict__ D, const float* __restrict__ stats,
                     float* __restrict__ part) {
    __shared__ float sm[256];
    int blk = blockIdx.x;
    int i = blk & 255, g = blk >> 8;
    int j = threadIdx.x;
    size_t idx = ((size_t)g) * LL + i * 256 + j;
    float v = siluf((D[idx] - stats[g * 2]) * stats[g * 2 + 1]);
    D[idx] = v;
    float s1 = blockReduceSum256(v, sm);
    if (j == 0) { part[(g * 256 + i) * 2] = s1; part[(g * 256 + i) * 2 + 1] = 0.f; }
}

// ---------------- SE MLP: mean -> silu(W1) -> sigmoid(W2) -> scale[b][c] ----------------
__global__ void k_se(const float* __restrict__ stats_h2, const float* __restrict__ W1,
                     const float* __restrict__ b1, const float* __restrict__ W2,
                     const float* __restrict__ b2, float* __restrict__ scale) {
    __shared__ float mh[64], sq[16];
    int b = blockIdx.x, t = threadIdx.x;
    mh[t] = stats_h2[(b * 64 + t) * 2];
    __syncthreads();
    if (t < 16) {
        float acc = b1[t];
        for (int c = 0; c < 64; c++) acc += mh[c] * W1[t * 64 + c];
        sq[t] = siluf(acc);
    }
    __syncthreads();
    float acc = b2[t];
    for (int s = 0; s < 16; s++) acc += sq[s] * W2[t * 16 + s];
    scale[b * 64 + t] = 1.f / (1.f + __expf(-acc));
}

// ---------------- pointwise project (64->32) with SE scaling, channel-last out ----------------
__global__ void k_pw(const float* __restrict__ H2, const float* __restrict__ scale,
                     const float* __restrict__ W_pw, float* __restrict__ HP) {
    __shared__ float w[2048]; __shared__ float sc[64];
    int b = blockIdx.x >> 8, i = blockIdx.x & 255;
    int t = threadIdx.x;
    for (int k = t; k < 2048; k += 256) w[k] = W_pw[k];
    if (t < 64) sc[t] = scale[b * 64 + t];
    __syncthreads();
    float acc[32];
#pragma unroll
    for (int p = 0; p < 32; p++) acc[p] = 0.f;
    for (int c = 0; c < 64; c++) {
        float hv = H2[((size_t)(b * 64 + c)) * LL + i * 256 + t] * sc[c];
#pragma unroll
        for (int p = 0; p < 32; p++) acc[p] += hv * w[p * 64 + c];
    }
    size_t base = (((size_t)(b * 256 + i)) * 256 + t) * 32;
#pragma unroll
    for (int p = 0; p < 32; p++) HP[base + p] = acc[p];
}

// ---------------- partial sums over last-dim-channel tensor for HP inorm ----------------
__global__ void k_part_lastdim(const float* __restrict__ HP, float* __restrict__ part) {
    __shared__ float sm[256];
    int blk = blockIdx.x;
    int i = blk & 255, p = (blk >> 8) & 31, b = blk >> 13;
    int j = threadIdx.x;
    float v = HP[(((size_t)(b * 256 + i)) * 256 + j) * 32 + p];
    float s1 = blockReduceSum256(v, sm);
    float s2 = blockReduceSum256(v * v, sm);
    if (j == 0) { part[((b * 32 + p) * 256 + i) * 2] = s1; part[((b * 32 + p) * 256 + i) * 2 + 1] = s2; }
}

// ---------------- final pair = pair1 + pn + inorm(HP); pn2=LN(pair); bias = pn2@W_bias ----------------
__global__ void k_final_pair(const float* __restrict__ pair1, const float* __restrict__ pn,
                             const float* __restrict__ HP, const float* __restrict__ stats_hp,
                             const float* __restrict__ g_pair, const float* __restrict__ b_pair,
                             const float* __restrict__ W_bias, float* __restrict__ pair_out,
                             float* __restrict__ biasb) {
    __shared__ float gp[32], bp[32], wb[128], mh[32], ih[32];
    int b = blockIdx.x >> 8, i = blockIdx.x & 255;
    int t = threadIdx.x;
    if (t < 32) {
        gp[t] = g_pair[t]; bp[t] = b_pair[t];
        mh[t] = stats_hp[(b * 32 + t) * 2]; ih[t] = stats_hp[(b * 32 + t) * 2 + 1];
    }
    if (t >= 32 && t < 160) wb[t - 32] = W_bias[t - 32];
    __syncthreads();
    size_t base = (((size_t)(b * 256 + i)) * 256 + t) * 32;
    float v[32]; float s1 = 0.f;
#pragma unroll
    for (int p = 0; p < 32; p++) {
        float pv = pair1[base + p] + pn[base + p] + (HP[base + p] - mh[p]) * ih[p];
        v[p] = pv; s1 += pv;
    }
    float m = s1 / 32.f, s2 = 0.f;
#pragma unroll
    for (int p = 0; p < 32; p++) { float d = v[p] - m; s2 += d * d; }
    float inv = rsqrtf(s2 / 32.f + 1e-5f);
    float bh[4] = {0.f, 0.f, 0.f, 0.f};
#pragma unroll
    for (int p = 0; p < 32; p++) {
        pair_out[base + p] = v[p];
        float pv2 = (v[p] - m) * inv * gp[p] + bp[p];
#pragma unroll
        for (int h = 0; h < 4; h++) bh[h] += pv2 * wb[p * 4 + h];
    }
#pragma unroll
    for (int h = 0; h < 4; h++) biasb[((size_t)(b * 4 + h)) * LL + i * 256 + t] = bh[h];
}

// ---------------- qkv = x @ W_qkv, transposed [B,H,L,HD] layout ----------------
__global__ void k_qkv(const float* __restrict__ xbuf, const float* __restrict__ W_qkv,
                      float* __restrict__ qT, float* __restrict__ kT, float* __restrict__ vT) {
    __shared__ float xs[64];
    int row = blockIdx.x, t = threadIdx.x;
    if (t < 64) xs[t] = xbuf[row * 64 + t];
    __syncthreads();
    if (t < 192) {
        float acc = 0.f;
        for (int k = 0; k < 64; k++) acc += xs[k] * W_qkv[k * 192 + t];
        int which = t / 64, rem = t % 64, h = rem / 16, dd = rem % 16;
        int b = row >> 8, l = row & 255;
        float* dst = which == 0 ? qT : (which == 1 ? kT : vT);
        dst[(((b * 4 + h) * 256) + l) * 16 + dd] = acc;
    }
}

// ---------------- attention: one block per (b,h,i) row ----------------
__global__ void k_attn(const float* __restrict__ qT, const float* __restrict__ kT,
                       const float* __restrict__ vT, const float* __restrict__ biasb,
                       float* __restrict__ ctxb) {
    __shared__ float q[16]; __shared__ float sc[256]; __shared__ float sm[256];
    int blk = blockIdx.x;
    int i = blk & 255, bh = blk >> 8;
    int t = threadIdx.x;
    if (t < 16) q[t] = qT[(bh * 256 + i) * 16 + t];
    __syncthreads();
    const float* kr = kT + (bh * 256 + t) * 16;
    float dot = 0.f;
#pragma unroll
    for (int d = 0; d < 16; d++) dot += q[d] * kr[d];
    float s = dot * 0.25f + biasb[((size_t)bh) * LL + i * 256 + t];
    sm[t] = s; __syncthreads();
    for (int st = 128; st > 0; st >>= 1) { if (t < st) sm[t] = fmaxf(sm[t], sm[t + st]); __syncthreads(); }
    float mx = sm[0]; __syncthreads();
    float w = __expf(s - mx);
    sc[t] = w;
    sm[t] = w; __syncthreads();
    for (int st = 128; st > 0; st >>= 1) { if (t < st) sm[t] += sm[t + st]; __syncthreads(); }
    float denom = sm[0]; __syncthreads();
    if (t < 16) {
        float acc = 0.f;
        for (int j = 0; j < 256; j++) acc += sc[j] * vT[(bh * 256 + j) * 16 + t];
        int b = bh >> 2, h = bh & 3;
        ctxb[((size_t)(b * 256 + i)) * 64 + h * 16 + t] = acc / denom;
    }
}

// ---------------- seq output projection + residual ----------------
__global__ void k_seq_out(const float* __restrict__ seq, const float* __restrict__ ctxb,
                          const float* __restrict__ W_outp, const float* __restrict__ b_outp,
                          float* __restrict__ out) {
    __shared__ float cs[64];
    int row = blockIdx.x, t = threadIdx.x;
    cs[t] = ctxb[row * 64 + t];
    __syncthreads();
    float acc = b_outp[t];
    for (int k = 0; k < 64; k++) acc += cs[k] * W_outp[k * 64 + t];
    out[row * 64 + t] = seq[row * 64 + t] + acc;
}

extern "C" void kernel_launch(void* const* d_in, const int* in_sizes, int n_in,
                              void* d_out, int out_size, void* d_ws, size_t ws_size,
                              hipStream_t stream) {
    const float* seq_repr   = (const float*)d_in[0];
    const float* pair_repr  = (const float*)d_in[1];
    const float* g_attn     = (const float*)d_in[2];
    const float* b_attn     = (const float*)d_in[3];
    const float* g_pair     = (const float*)d_in[4];
    const float* b_pair     = (const float*)d_in[5];
    const float* W_opm_lin  = (const float*)d_in[6];
    const float* b_opm_lin  = (const float*)d_in[7];
    const float* g_opm      = (const float*)d_in[8];
    const float* b_opm      = (const float*)d_in[9];
    const float* W_opm_proj = (const float*)d_in[10];
    const float* b_opm_proj = (const float*)d_in[11];
    const float* W_exp      = (const float*)d_in[12];
    const float* W_dw       = (const float*)d_in[13];
    const float* W_se1      = (const float*)d_in[14];
    const float* b_se1      = (const float*)d_in[15];
    const float* W_se2      = (const float*)d_in[16];
    const float* b_se2      = (const float*)d_in[17];
    const float* W_pw       = (const float*)d_in[18];
    const float* W_qkv      = (const float*)d_in[19];
    const float* W_outp     = (const float*)d_in[20];
    const float* b_outp     = (const float*)d_in[21];
    const float* W_bias     = (const float*)d_in[22];

    float* out_seq  = (float*)d_out;            // [2,256,64]
    float* out_pair = out_seq + BB * LS * SD;   // [2,256,256,32]

    float* W = (float*)d_ws;
    size_t o = 0;
    auto alloc = [&](size_t n) { float* p = W + o; o += n; return p; };
    float* xbuf     = alloc(512 * 64);
    float* abuf     = alloc(512 * 32);
    float* mstat    = alloc(512);
    float* sstat    = alloc(512);
    float* G        = alloc(32768);
    float* C1       = alloc(32);
    float* C2       = alloc(32);
    float* Y2       = alloc((size_t)BB * 32 * 8192);
    float* pair1    = alloc((size_t)BB * LS * LS * PD);
    float* pnb      = alloc((size_t)BB * LS * LS * PD);
    float* Ebuf     = alloc((size_t)BB * EXPC * LL);
    float* Dbuf     = alloc((size_t)BB * EXPC * LL);
    float* HP       = alloc((size_t)BB * LS * LS * PD);
    float* biasb    = alloc((size_t)BB * NH * LL);
    float* qT       = alloc((size_t)BB * NH * LS * HD);
    float* kT       = alloc((size_t)BB * NH * LS * HD);
    float* vT       = alloc((size_t)BB * NH * LS * HD);
    float* ctxb     = alloc(512 * 64);
    float* part_e   = alloc((size_t)BB * EXPC * LS * 2);
    float* stats_e  = alloc(BB * EXPC * 2);
    float* part_d   = alloc((size_t)BB * EXPC * LS * 2);
    float* stats_d  = alloc(BB * EXPC * 2);
    float* part_h2  = alloc((size_t)BB * EXPC * LS * 2);
    float* stats_h2 = alloc(BB * EXPC * 2);
    float* scale_se = alloc(BB * EXPC);
    float* part_hp  = alloc((size_t)BB * PD * LS * 2);
    float* stats_hp = alloc(BB * PD * 2);

    // 1. seq LN, a, stats
    k_seq_ln_a<<<512, 64, 0, stream>>>(seq_repr, g_attn, b_attn, W_opm_lin, b_opm_lin,
                                       xbuf, abuf, mstat, sstat);
    // 2. G, C1, C2
    k_prep_G<<<129, 256, 0, stream>>>(g_opm, b_opm, W_opm_proj, b_opm_proj, G, C1, C2);
    // 3. Y
    k_Y<<<512, 1024, 0, stream>>>(abuf, G, Y2);
    // 4. WMMA GEMM + fused LN + pair residual
    k_opm_wmma<<<4096, 128, 0, stream>>>(abuf, Y2, mstat, sstat, C1, C2, pair_repr, pair1);
    // 5. pn = LN(pair1)
    k_pn<<<512, 256, 0, stream>>>(pair1, g_pair, b_pair, pnb);
    // 6. expand 1x1
    k_e<<<BB * EXPC * LS, 256, 0, stream>>>(pnb, W_exp, Ebuf, part_e);
    k_reduce_stats<<<BB * EXPC, 256, 0, stream>>>(part_e, stats_e, 1.f / 65536.f);
    k_silu_norm<<<BB * EXPC * LL / 256, 256, 0, stream>>>(Ebuf, stats_e);
    // 7. depthwise 3x3
    k_dw<<<BB * EXPC * LS, 256, 0, stream>>>(Ebuf, W_dw, Dbuf, part_d);
    k_reduce_stats<<<BB * EXPC, 256, 0, stream>>>(part_d, stats_d, 1.f / 65536.f);
    k_h2<<<BB * EXPC * LS, 256, 0, stream>>>(Dbuf, stats_d, part_h2);
    k_reduce_stats<<<BB * EXPC, 256, 0, stream>>>(part_h2, stats_h2, 1.f / 65536.f);
    // 8. SE
    k_se<<<BB, 64, 0, stream>>>(stats_h2, W_se1, b_se1, W_se2, b_se2, scale_se);
    // 9. pointwise project
    k_pw<<<BB * LS, 256, 0, stream>>>(Dbuf, scale_se, W_pw, HP);
    k_part_lastdim<<<BB * PD * LS, 256, 0, stream>>>(HP, part_hp);
    k_reduce_stats<<<BB * PD, 256, 0, stream>>>(part_hp, stats_hp, 1.f / 65536.f);
    // 10. final pair + attention bias
    k_final_pair<<<BB * LS, 256, 0, stream>>>(pair1, pnb, HP, stats_hp, g_pair, b_pair,
                                              W_bias, out_pair, biasb);
    // 11. attention
    k_qkv<<<512, 256, 0, stream>>>(xbuf, W_qkv, qT, kT, vT);
    k_attn<<<BB * NH * LS, 256, 0, stream>>>(qT, kT, vT, biasb, ctxb);
    k_seq_out<<<512, 64, 0, stream>>>(seq_repr, ctxb, W_outp, b_outp, out_seq);
    (void)in_sizes; (void)n_in; (void)out_size; (void)ws_size;
}